// HoloLayer_30253749633409
// MI455X (gfx1250) — compile-verified
//
#include <hip/hip_runtime.h>
#include <hip/hip_bf16.h>
#include <math.h>

// Problem constants (from reference): B=4, S=4096, D=1024
#define BB 4
#define SS 4096
#define DD 1024
#define BS (BB * SS)           // 16384 rows
#define NCH 16                 // scan chunks along S
#define CHL (SS / NCH)         // 256 per chunk

typedef __attribute__((ext_vector_type(2))) float v2f;
typedef __attribute__((ext_vector_type(8))) float v8f;

// ---------------------------------------------------------------------------
// GEMM: out[m,n] = sum_k X[m,k] * W[n,k] + bias[n]
// One wave computes a 64x64 output tile as a 4x4 grid of 16x16 WMMA tiles
// (V_WMMA_F32_16X16X4_F32). Register tiling: 8 fragment loads feed 16 WMMAs
// per K-step. __launch_bounds__(256,1) relaxes the VGPR budget so the 16
// accumulators (128 VGPRs) stay resident; row sub-tiles are addressed via
// immediate offsets (i*16*K*4 = 64KiB < 2^23) off a single base pointer.
// A fragment (16x4, MxK): lane<16 -> row=lane&15, K pair {k0,k0+1};
//                         lane>=16 -> K pair {k0+2,k0+3}.
// B fragment (4x16, KxN): mirrored with N instead of M (W row-major = B^T).
// D/C (16x16 f32): vgpr g, lane<16 -> (m=g, n=lane); lane>=16 -> (m=8+g, n=lane-16)
// ---------------------------------------------------------------------------
__global__ void __launch_bounds__(256, 1)
gemm_bias_wmma(const float* __restrict__ X,
               const float* __restrict__ W,
               const float* __restrict__ bias,
               float* __restrict__ out) {
  constexpr int N = DD, K = DD;
  const int wave = threadIdx.x >> 5;
  const int lane = threadIdx.x & 31;
  const int wid = blockIdx.x * 8 + wave;   // wave-linear 64x64 tile id
  const int wtn = N / 64;                  // 16 tiles across N
  const int wm = wid / wtn;
  const int wn = wid % wtn;
  const int l16 = lane & 15;
  const int half = lane >> 4;

  // Single base pointer per matrix; sub-tile rows via immediate offsets.
  const float* aBase = X + (size_t)(wm * 64 + l16) * K + 2 * half;
  const float* bBase = W + (size_t)(wn * 64 + l16) * K + 2 * half;

  v8f acc[4][4];
  const v8f zero = {};
#pragma unroll
  for (int mi = 0; mi < 4; ++mi)
#pragma unroll
    for (int ni = 0; ni < 4; ++ni) acc[mi][ni] = zero;

  for (int k = 0; k < K; k += 4) {
    v2f a[4], b[4];
#pragma unroll
    for (int i = 0; i < 4; ++i) {
      a[i] = *(const v2f*)(aBase + i * (16 * K) + k);  // +i*64KiB immediate
      b[i] = *(const v2f*)(bBase + i * (16 * K) + k);
    }
#pragma unroll
    for (int mi = 0; mi < 4; ++mi)
#pragma unroll
      for (int ni = 0; ni < 4; ++ni)
        acc[mi][ni] = __builtin_amdgcn_wmma_f32_16x16x4_f32(
            /*neg_a=*/false, a[mi], /*neg_b=*/false, b[ni],
            /*c_mod=*/(short)0, acc[mi][ni],
            /*reuse_a=*/false, /*reuse_b=*/false);
  }

#pragma unroll
  for (int ni = 0; ni < 4; ++ni) {
    const float bv = bias[wn * 64 + ni * 16 + l16];
#pragma unroll
    for (int mi = 0; mi < 4; ++mi) {
      float* orow = out + (size_t)(wm * 64 + mi * 16 + half * 8) * N +
                    wn * 64 + ni * 16 + l16;
#pragma unroll
      for (int g = 0; g < 8; ++g) orow[(size_t)g * N] = acc[mi][ni][g] + bv;
    }
  }
}

// ---------------------------------------------------------------------------
// 1024-point radix-2 complex FFT in LDS. 256 threads, 512 butterflies/stage.
// sign = -1 forward (jnp.fft.fft), +1 inverse (scale 1/N applied by caller).
// Caller must have synced after filling `s` in bit-reversed order.
// ---------------------------------------------------------------------------
__device__ __forceinline__ void fft1024(float2* s, int tid, float sign) {
  for (int len = 2; len <= 1024; len <<= 1) {
    const int half = len >> 1;
    for (int bb = tid; bb < 512; bb += 256) {
      const int j = bb & (half - 1);
      const int i0 = 2 * bb - j;
      const int i1 = i0 + half;
      const float ang = sign * 6.28318530717958647692f * (float)j / (float)len;
      float sn, cs;
      __sincosf(ang, &sn, &cs);
      const float2 u = s[i0];
      const float2 t = s[i1];
      const float wtx = cs * t.x - sn * t.y;
      const float wty = cs * t.y + sn * t.x;
      s[i0] = make_float2(u.x + wtx, u.y + wty);
      s[i1] = make_float2(u.x - wtx, u.y - wty);
    }
    __syncthreads();
  }
}

__device__ __forceinline__ int rev10(int i) { return (int)(__brev((unsigned)i) >> 22); }

// ---------------------------------------------------------------------------
// Per (b,s) row: FFT(k), FFT(v), FFT(q); p = K^*V^ (elementwise complex),
// qc = conj(Q^).  One block per row.
// ---------------------------------------------------------------------------
__global__ void fft_combine(const float* __restrict__ Kr,
                            const float* __restrict__ Vr,
                            const float* __restrict__ Qr,
                            float2* __restrict__ p,
                            float2* __restrict__ qc) {
  __shared__ float2 sh[1024];
  const int row = blockIdx.x;
  const int tid = threadIdx.x;
  const size_t base = (size_t)row * DD;

  // FFT of k
#pragma unroll
  for (int t = 0; t < 4; ++t) {
    const int i = tid + (t << 8);
    sh[rev10(i)] = make_float2(Kr[base + i], 0.f);
  }
  __syncthreads();
  fft1024(sh, tid, -1.f);
  float2 khat[4];
#pragma unroll
  for (int t = 0; t < 4; ++t) khat[t] = sh[tid + (t << 8)];
  __syncthreads();

  // FFT of v, then p = khat * vhat
#pragma unroll
  for (int t = 0; t < 4; ++t) {
    const int i = tid + (t << 8);
    sh[rev10(i)] = make_float2(Vr[base + i], 0.f);
  }
  __syncthreads();
  fft1024(sh, tid, -1.f);
#pragma unroll
  for (int t = 0; t < 4; ++t) {
    const int i = tid + (t << 8);
    const float2 a = khat[t];
    const float2 b = sh[i];
    p[base + i] = make_float2(a.x * b.x - a.y * b.y, a.x * b.y + a.y * b.x);
  }
  __syncthreads();

  // FFT of q, store conj
#pragma unroll
  for (int t = 0; t < 4; ++t) {
    const int i = tid + (t << 8);
    sh[rev10(i)] = make_float2(Qr[base + i], 0.f);
  }
  __syncthreads();
  fft1024(sh, tid, -1.f);
#pragma unroll
  for (int t = 0; t < 4; ++t) {
    const int i = tid + (t << 8);
    const float2 q = sh[i];
    qc[base + i] = make_float2(q.x, -q.y);
  }
}

// ---------------------------------------------------------------------------
// Causal scan over S, hierarchical: chunk sums -> exclusive chunk scan ->
// in-chunk cumsum fused with conj(q) multiply (in-place over p).
// ---------------------------------------------------------------------------
__global__ void chunk_sum(const float2* __restrict__ p, float2* __restrict__ ch) {
  const int t = blockIdx.x * 256 + threadIdx.x;  // < B*NCH*D
  const int d = t & (DD - 1);
  const int c = (t >> 10) & (NCH - 1);
  const int b = t >> 14;
  float2 acc = make_float2(0.f, 0.f);
  size_t idx = ((size_t)b * SS + (size_t)c * CHL) * DD + d;
  for (int s = 0; s < CHL; ++s, idx += DD) {
    const float2 v = p[idx];
    acc.x += v.x;
    acc.y += v.y;
  }
  ch[t] = acc;  // layout [b][c][d] == t
}

__global__ void chunk_scan(float2* __restrict__ ch) {
  const int t = blockIdx.x * 256 + threadIdx.x;  // < B*D
  const int d = t & (DD - 1);
  const int b = t >> 10;
  float2 run = make_float2(0.f, 0.f);
  size_t idx = (size_t)b * NCH * DD + d;
  for (int c = 0; c < NCH; ++c, idx += DD) {
    const float2 v = ch[idx];
    ch[idx] = run;  // exclusive prefix
    run.x += v.x;
    run.y += v.y;
  }
}

__global__ void scan_apply(float2* __restrict__ p,
                           const float2* __restrict__ qc,
                           const float2* __restrict__ ch) {
  const int t = blockIdx.x * 256 + threadIdx.x;  // < B*NCH*D
  const int d = t & (DD - 1);
  const int c = (t >> 10) & (NCH - 1);
  const int b = t >> 14;
  float2 acc = ch[t];
  size_t idx = ((size_t)b * SS + (size_t)c * CHL) * DD + d;
  for (int s = 0; s < CHL; ++s, idx += DD) {
    const float2 v = p[idx];
    acc.x += v.x;
    acc.y += v.y;
    const float2 q = qc[idx];
    p[idx] = make_float2(acc.x * q.x - acc.y * q.y, acc.x * q.y + acc.y * q.x);
  }
}

// ---------------------------------------------------------------------------
// Per (b,s) row: L2 norm over D, scale by 1/(1e-9+nrm), inverse FFT,
// real part / sqrt(D) -> out. ifft scale 1/1024 folded with 1/32.
// ---------------------------------------------------------------------------
__global__ void norm_ifft(const float2* __restrict__ qd, float* __restrict__ out) {
  __shared__ float2 sh[1024];
  __shared__ float red[256];
  const int row = blockIdx.x;
  const int tid = threadIdx.x;
  const size_t base = (size_t)row * DD;

  float2 r[4];
  float ps = 0.f;
#pragma unroll
  for (int t = 0; t < 4; ++t) {
    r[t] = qd[base + tid + (t << 8)];
    ps += r[t].x * r[t].x + r[t].y * r[t].y;
  }
  red[tid] = ps;
  __syncthreads();
  for (int off = 128; off > 0; off >>= 1) {
    if (tid < off) red[tid] += red[tid + off];
    __syncthreads();
  }
  const float scale = 1.f / (1e-9f + sqrtf(red[0]));

#pragma unroll
  for (int t = 0; t < 4; ++t) {
    const int i = tid + (t << 8);
    sh[rev10(i)] = make_float2(r[t].x * scale, r[t].y * scale);
  }
  __syncthreads();
  fft1024(sh, tid, +1.f);  // inverse direction

  const float k = 1.f / (1024.f * 32.f);  // (1/N) * (1/sqrt(D))
#pragma unroll
  for (int t = 0; t < 4; ++t) {
    const int i = tid + (t << 8);
    out[base + i] = sh[i].x * k;
  }
}

// ---------------------------------------------------------------------------
extern "C" void kernel_launch(void* const* d_in, const int* in_sizes, int n_in,
                              void* d_out, int out_size, void* d_ws, size_t ws_size,
                              hipStream_t stream) {
  const float* x  = (const float*)d_in[0];
  const float* Wk = (const float*)d_in[1];
  const float* bk = (const float*)d_in[2];
  const float* Wv = (const float*)d_in[3];
  const float* bv = (const float*)d_in[4];
  const float* Wq = (const float*)d_in[5];
  const float* bq = (const float*)d_in[6];
  float* out = (float*)d_out;

  char* ws = (char*)d_ws;
  const size_t szReal = (size_t)BS * DD * sizeof(float);    // 64 MB
  const size_t szCplx = (size_t)BS * DD * sizeof(float2);   // 128 MB
  float*  Kr = (float*)(ws);
  float*  Vr = (float*)(ws + szReal);
  float*  Qr = (float*)(ws + 2 * szReal);
  float2* p  = (float2*)(ws + 3 * szReal);
  float2* qc = (float2*)(ws + 3 * szReal + szCplx);
  float2* ch = (float2*)(ws + 3 * szReal + 2 * szCplx);     // B*NCH*D complex

  // 64x64 tile per wave: (16384/64)*(1024/64) = 4096 waves, 8 waves/block.
  const int wavesTotal = (BS / 64) * (DD / 64);
  const int gemmBlocks = wavesTotal / 8;  // 512

  gemm_bias_wmma<<<gemmBlocks, 256, 0, stream>>>(x, Wk, bk, Kr);
  gemm_bias_wmma<<<gemmBlocks, 256, 0, stream>>>(x, Wv, bv, Vr);
  gemm_bias_wmma<<<gemmBlocks, 256, 0, stream>>>(x, Wq, bq, Qr);

  fft_combine<<<BS, 256, 0, stream>>>(Kr, Vr, Qr, p, qc);

  chunk_sum<<<(BB * NCH * DD) / 256, 256, 0, stream>>>(p, ch);
  chunk_scan<<<(BB * DD) / 256, 256, 0, stream>>>(ch);
  scan_apply<<<(BB * NCH * DD) / 256, 256, 0, stream>>>(p, qc, ch);

  norm_ifft<<<BS, 256, 0, stream>>>(p, out);
}